// EncoderGPECls_21397527069086
// MI455X (gfx1250) — compile-verified
//
#include <hip/hip_runtime.h>
#include <math.h>

#define BATCH 8
#define NPTS  4096
#define KNN   16

typedef float v2f __attribute__((ext_vector_type(2)));
typedef float v8f __attribute__((ext_vector_type(8)));

// ---------------------------------------------------------------------------
// Kernel 1: kNN via V_WMMA_F32_16X16X4_F32 distance tiles.
// Block = 256 threads = 8 waves; each wave owns 32 query points (two A tiles).
// Per iteration: 16 candidates (one B tile), two WMMAs -> 32x16 d2 tile in
// wave-private LDS; lane L owns query row L -> no cross-lane merge needed.
// All LDS exchange is intra-wave: DS ops are in-order per wave on CDNA5, so
// only a compiler fence (wave_barrier) is required -- no s_barrier traffic.
// Grid = BATCH * (NPTS/256) blocks.
// ---------------------------------------------------------------------------
#define ROWSTRIDE 20  // floats; 80B row pitch: 16B-aligned for float4 reads,
                      // spreads rows across the 64 LDS banks

__global__ __launch_bounds__(256) void knn_wmma_kernel(const float* __restrict__ xyz,
                                                       int* __restrict__ knn_idx) {
  __shared__ float s_d2[8][32 * ROWSTRIDE];   // per-wave 32x16 tile (padded)

  const int wave = threadIdx.x >> 5;
  const int lane = threadIdx.x & 31;
  const int half = lane >> 4;          // 0: lanes 0-15, 1: lanes 16-31
  const int l15  = lane & 15;

  const int b  = blockIdx.x >> 4;                       // 16 blocks per batch
  const int q0 = ((blockIdx.x & 15) << 8) + (wave << 5);
  const float* __restrict__ base = xyz + (size_t)b * NPTS * 3;
  float* __restrict__ tile = &s_d2[wave][0];

  // Queries for the two A tiles (rows 0-15 and 16-31 of the d2 tile).
  const int qa = q0 + l15;
  const int qb = q0 + 16 + l15;
  const float ax = base[qa*3+0], ay = base[qa*3+1], az = base[qa*3+2];
  const float bx = base[qb*3+0], by = base[qb*3+1], bz = base[qb*3+2];
  const float sqQa = ax*ax + ay*ay + az*az;
  const float sqQb = bx*bx + by*by + bz*bz;

  // A layout (16x4, K=3 zero-padded): lanes 0-15 hold K=0/K=1, 16-31 K=2/K=3.
  v2f A0, A1;
  A0.x = half ? az : ax;  A0.y = half ? 0.0f : ay;
  A1.x = half ? bz : bx;  A1.y = half ? 0.0f : by;

  // |query m|^2 for C-tile rows m = v + 8*half (tile0) / 16+v+8*half (tile1).
  float sqA0[8], sqA1[8];
  #pragma unroll
  for (int v = 0; v < 8; ++v) {
    sqA0[v] = __shfl(sqQa, v + 8*half, 32);
    sqA1[v] = __shfl(sqQb, v + 8*half, 32);
  }

  // This lane's selection query = row `lane` of the tile.
  const int q = q0 + lane;

  // Sorted ascending top-16 (distance, index) in registers.
  float ld[KNN]; int li[KNN];
  #pragma unroll
  for (int k = 0; k < KNN; ++k) { ld[k] = 3.0e38f; li[k] = 0; }

  for (int j0 = 0; j0 < NPTS; j0 += 16) {
    // Candidate tile: both halves load the same 16 candidates (K-split rows).
    const int c = j0 + l15;
    const float cx = base[c*3+0];
    const float cy = base[c*3+1];
    const float cz = base[c*3+2];
    const float sqC = cx*cx + cy*cy + cz*cz;
    if (j0 + 64 < NPTS)
      __builtin_prefetch(base + (size_t)(j0 + 64) * 3, 0, 1);

    v2f Bm;
    Bm.x = half ? cz : cx;
    Bm.y = half ? 0.0f : cy;

    v8f acc0 = {}, acc1 = {};
    acc0 = __builtin_amdgcn_wmma_f32_16x16x4_f32(false, A0, false, Bm,
                                                 (short)0, acc0, false, false);
    acc1 = __builtin_amdgcn_wmma_f32_16x16x4_f32(false, A1, false, Bm,
                                                 (short)0, acc1, false, false);

    // d2 tile -> LDS: col = l15, rows v+8*half (tile0) and 16+v+8*half (tile1).
    #pragma unroll
    for (int v = 0; v < 8; ++v) {
      const int m = v + 8*half;
      tile[m * ROWSTRIDE + l15]        = sqA0[v] + sqC - 2.0f * acc0[v];
      tile[(16 + m) * ROWSTRIDE + l15] = sqA1[v] + sqC - 2.0f * acc1[v];
    }
    __builtin_amdgcn_wave_barrier();   // intra-wave LDS exchange: compiler fence only

    // Selection: lane scans its own row (16 candidates) as 4x float4.
    const float4* __restrict__ row =
        reinterpret_cast<const float4*>(tile + lane * ROWSTRIDE);
    #pragma unroll
    for (int c4 = 0; c4 < 4; ++c4) {
      const float4 dv = row[c4];
      const float dd[4] = {dv.x, dv.y, dv.z, dv.w};
      #pragma unroll
      for (int e = 0; e < 4; ++e) {
        const int   cj = j0 + c4*4 + e;
        const float d  = dd[e];
        if (cj != q && d < ld[KNN-1]) {
          // Branchless cndmask cascade (flat under divergence).
          float cd = d; int ci = cj;
          #pragma unroll
          for (int t = 0; t < KNN; ++t) {
            const bool  lt = cd < ld[t];
            const float f0 = lt ? cd    : ld[t];
            const float f1 = lt ? ld[t] : cd;
            const int   i0 = lt ? ci    : li[t];
            const int   i1 = lt ? li[t] : ci;
            ld[t] = f0; li[t] = i0; cd = f1; ci = i1;
          }
        }
      }
    }
    __builtin_amdgcn_wave_barrier();   // WAR fence before next tile's stores
  }

  // Each lane owns a complete sorted top-16: store as 4x b128.
  int4* __restrict__ outp =
      reinterpret_cast<int4*>(knn_idx + ((size_t)b * NPTS + q) * KNN);
  outp[0] = make_int4(li[0],  li[1],  li[2],  li[3]);
  outp[1] = make_int4(li[4],  li[5],  li[6],  li[7]);
  outp[2] = make_int4(li[8],  li[9],  li[10], li[11]);
  outp[3] = make_int4(li[12], li[13], li[14], li[15]);
}

// ---------------------------------------------------------------------------
// Kernel 2: per-point neighbor stats: 3x3 covariance -> min eigval curvature,
// and per-dim local std (= sqrt(diag(cov)) since ddof and centering match).
// ---------------------------------------------------------------------------
__global__ __launch_bounds__(256) void stats_kernel(const float* __restrict__ xyz,
                                                    const int* __restrict__ knn_idx,
                                                    float* __restrict__ curv,
                                                    float* __restrict__ lstd) {
  const int gid = blockIdx.x * blockDim.x + threadIdx.x;
  if (gid >= BATCH * NPTS) return;
  const int b = gid >> 12;
  const float* __restrict__ base = xyz + (size_t)b * NPTS * 3;
  const int* __restrict__ nb = knn_idx + (size_t)gid * KNN;

  float px[KNN], py[KNN], pz[KNN];
  float sx = 0.0f, sy = 0.0f, sz = 0.0f;
  #pragma unroll
  for (int k = 0; k < KNN; ++k) {
    const int j = nb[k];
    px[k] = base[j*3+0]; py[k] = base[j*3+1]; pz[k] = base[j*3+2];
    sx += px[k]; sy += py[k]; sz += pz[k];
  }
  const float inv = 1.0f / (float)KNN;
  const float mx = sx*inv, my = sy*inv, mz = sz*inv;

  float cxx=0,cyy=0,czz=0,cxy=0,cxz=0,cyz=0;
  #pragma unroll
  for (int k = 0; k < KNN; ++k) {
    const float dx = px[k]-mx, dy = py[k]-my, dz = pz[k]-mz;
    cxx += dx*dx; cyy += dy*dy; czz += dz*dz;
    cxy += dx*dy; cxz += dx*dz; cyz += dy*dz;
  }
  const float s15 = 1.0f / (float)(KNN - 1);
  cxx*=s15; cyy*=s15; czz*=s15; cxy*=s15; cxz*=s15; cyz*=s15;

  // Closed-form smallest eigenvalue of symmetric 3x3.
  const float p1 = cxy*cxy + cxz*cxz + cyz*cyz;
  const float tr = cxx + cyy + czz;
  const float qm = tr * (1.0f/3.0f);
  const float p2 = (cxx-qm)*(cxx-qm) + (cyy-qm)*(cyy-qm) + (czz-qm)*(czz-qm) + 2.0f*p1;
  const float p  = sqrtf(p2 * (1.0f/6.0f));
  float emin;
  if (p < 1e-12f) {
    emin = qm;
  } else {
    const float ip = 1.0f / p;
    const float b00=(cxx-qm)*ip, b11=(cyy-qm)*ip, b22=(czz-qm)*ip;
    const float b01=cxy*ip, b02=cxz*ip, b12=cyz*ip;
    float detB = b00*(b11*b22 - b12*b12) - b01*(b01*b22 - b12*b02)
               + b02*(b01*b12 - b11*b02);
    float r = 0.5f * detB;
    r = fminf(1.0f, fmaxf(-1.0f, r));
    const float phi = acosf(r) * (1.0f/3.0f);
    emin = qm + 2.0f*p*cosf(phi + 2.0943951023931953f); // + 2*pi/3 -> smallest
  }
  curv[gid] = emin / (tr + 1e-6f);
  lstd[(size_t)gid*3+0] = sqrtf(fmaxf(cxx, 0.0f));
  lstd[(size_t)gid*3+1] = sqrtf(fmaxf(cyy, 0.0f));
  lstd[(size_t)gid*3+2] = sqrtf(fmaxf(czz, 0.0f));
}

// ---------------------------------------------------------------------------
// Kernel 3: per-batch reductions: mean(curv) and per-(batch,dim) std of xyz.
// ---------------------------------------------------------------------------
__global__ __launch_bounds__(256) void reduce_kernel(const float* __restrict__ xyz,
                                                     const float* __restrict__ curv,
                                                     float* __restrict__ mean_curv,
                                                     float* __restrict__ std_bd) {
  __shared__ float red[256];
  const int b = blockIdx.x, t = threadIdx.x;
  float sc = 0.0f, sd0=0,sd1=0,sd2=0, sq0=0,sq1=0,sq2=0;
  for (int i = t; i < NPTS; i += 256) {
    sc += curv[b*NPTS + i];
    const float v0 = xyz[((size_t)b*NPTS + i)*3 + 0];
    const float v1 = xyz[((size_t)b*NPTS + i)*3 + 1];
    const float v2 = xyz[((size_t)b*NPTS + i)*3 + 2];
    sd0 += v0; sq0 += v0*v0;
    sd1 += v1; sq1 += v1*v1;
    sd2 += v2; sq2 += v2*v2;
  }
  float vals[7] = {sc, sd0, sd1, sd2, sq0, sq1, sq2};
  float tot[7];
  #pragma unroll
  for (int j = 0; j < 7; ++j) {
    red[t] = vals[j]; __syncthreads();
    for (int s = 128; s > 0; s >>= 1) { if (t < s) red[t] += red[t+s]; __syncthreads(); }
    tot[j] = red[0]; __syncthreads();
  }
  if (t == 0) {
    mean_curv[b] = tot[0] / (float)NPTS;
    #pragma unroll
    for (int d = 0; d < 3; ++d) {
      const float var = (tot[4+d] - tot[1+d]*tot[1+d]/(float)NPTS) / (float)(NPTS - 1);
      std_bd[b*3 + d] = sqrtf(fmaxf(var, 0.0f));
    }
  }
}

__global__ void gstd_kernel(const float* __restrict__ std_bd, float* __restrict__ gstd) {
  if (blockIdx.x == 0 && threadIdx.x == 0) {
    float s = 0.0f;
    for (int i = 0; i < BATCH*3; ++i) s += std_bd[i];
    *gstd = s * (1.0f / (float)(BATCH*3));
  }
}

// ---------------------------------------------------------------------------
// Kernel 4: final embedding. 128 threads per point (one per output channel),
// 2 points per 256-thread block.
// ---------------------------------------------------------------------------
__global__ __launch_bounds__(256) void encode_kernel(const float* __restrict__ xyz,
                                                     const float* __restrict__ curv,
                                                     const float* __restrict__ lstd,
                                                     const float* __restrict__ mean_curv,
                                                     const float* __restrict__ gstd_p,
                                                     float* __restrict__ out) {
  const int o  = threadIdx.x & 127;
  const int pt = (blockIdx.x << 1) + (threadIdx.x >> 7);
  const int b  = pt >> 12;

  const float gstd = *gstd_p;
  const float x0 = xyz[(size_t)pt*3 + 0];
  const float x1 = xyz[(size_t)pt*3 + 1];
  const float x2 = xyz[(size_t)pt*3 + 2];

  const float w      = 1.0f / (1.0f + expf(-10.0f * (curv[pt] - mean_curv[b])));
  const float asig_g = 0.3f * (1.0f + gstd) + 1e-6f;
  const float blend  = 1.0f / (1.0f + expf(-(gstd - 0.1f) * 10.0f));

  // OUT_IDX[o] = int(linspace(0,128,128)[o]) == (o*128)/127 exactly.
  const int idx = (o * 128) / 127;
  const int d   = idx / 43;
  const int j   = idx - 43 * d;
  const float fv = -1.0f + (float)(j + 1) * (1.0f / 22.0f);   // linspace(-1,1,45)[1:-1]
  const float xd = (d == 0) ? x0 : ((d == 1) ? x1 : x2);

  const float t  = (xd - fv) / asig_g;
  const float e1 = blend * expf(-0.5f*t*t) + (1.0f - blend) * cosf(t);

  const float ls = lstd[(size_t)pt*3 + d];
  const float t2 = (xd - fv) / (0.3f * (1.0f + ls) + 1e-6f);
  const float e2 = expf(-0.5f*t2*t2);

  out[(size_t)pt*128 + o] = w * e1 + (1.0f - w) * e2;
}

// ---------------------------------------------------------------------------
extern "C" void kernel_launch(void* const* d_in, const int* in_sizes, int n_in,
                              void* d_out, int out_size, void* d_ws, size_t ws_size,
                              hipStream_t stream) {
  const float* xyz = (const float*)d_in[0];
  float* out = (float*)d_out;

  char* ws = (char*)d_ws;
  int*   knn_idx   = (int*)ws;                                   // 8*4096*16*4B = 2 MiB
  float* curv      = (float*)(ws + (size_t)BATCH*NPTS*KNN*4);    // 128 KiB
  float* lstd      = curv + (size_t)BATCH*NPTS;                  // 384 KiB
  float* mean_curv = lstd + (size_t)BATCH*NPTS*3;                // 8 floats
  float* std_bd    = mean_curv + BATCH;                          // 24 floats
  float* gstd      = std_bd + BATCH*3;                           // 1 float

  knn_wmma_kernel<<<BATCH*(NPTS/256), 256, 0, stream>>>(xyz, knn_idx);
  stats_kernel<<<(BATCH*NPTS + 255)/256, 256, 0, stream>>>(xyz, knn_idx, curv, lstd);
  reduce_kernel<<<BATCH, 256, 0, stream>>>(xyz, curv, mean_curv, std_bd);
  gstd_kernel<<<1, 32, 0, stream>>>(std_bd, gstd);
  encode_kernel<<<BATCH*NPTS/2, 256, 0, stream>>>(xyz, curv, lstd, mean_curv, gstd, out);
}